// LIChI_55740085567820
// MI455X (gfx1250) — compile-verified
//
#include <hip/hip_runtime.h>

// ---------------------------------------------------------------------------
// LIChI denoiser for gfx1250 (MI455X). Fully fused per-group pipeline:
//   block-matching -> [gather->Gram(WMMA f32 16x16x4)->Cholesky->apply(WMMA)
//                      ->atomic aggregate] -> normalize, iterated 9x.
// All math in f32; matrix products on V_WMMA_F32_16X16X4_F32.
// ---------------------------------------------------------------------------

typedef __attribute__((ext_vector_type(2))) float v2f;
typedef __attribute__((ext_vector_type(8))) float v8f;

#define HH 128
#define WW 128
#define NPIX (HH*WW)

#define P1_ 11
#define K1_ 16
#define N1_ 121          // P1*P1
#define HP1 118          // H - P1 + 1
#define HR1 40           // ceil((HP1-1)/3)+1
#define B1  (HR1*HR1)

#define P2_ 6
#define K2_ 64
#define N2_ 36
#define HP2 123
#define HR2 42
#define B2  (HR2*HR2)

#define WIN_ 65
#define V_   32
#define EPS_ 1e-6f

static __device__ __forceinline__ int imin(int a, int b) { return a < b ? a : b; }

static __device__ __forceinline__ v8f wmma_f32_4(v2f a, v2f b, v8f c) {
  // D = A(16x4) * B(4x16) + C ; symmetric Gram passes a==b.
  return __builtin_amdgcn_wmma_f32_16x16x4_f32(false, a, false, b, (short)0, c,
                                               false, false);
}

// ---------------------------------------------------------------------------
// Block matching: one workgroup per reference patch. Distance table in LDS,
// then K rounds of argmin extraction.
// ---------------------------------------------------------------------------
template <int P, int K>
__global__ void __launch_bounds__(256)
bm_kernel(const float* __restrict__ img, int* __restrict__ idx_out, int hr) {
  const int HP = HH - P + 1;
  __shared__ float ref[P * P];
  __shared__ float dist[WIN_ * WIN_];
  __shared__ float rval[256];
  __shared__ int   ridx[256];

  const int b   = blockIdx.x;
  const int bi  = b / hr, bj = b % hr;
  const int rr  = imin(bi * 3, HP - 1);
  const int rc  = imin(bj * 3, HP - 1);
  const int tid = threadIdx.x;
  const float FINF = __builtin_inff();

  for (int e = tid; e < P * P; e += 256)
    ref[e] = img[(rr + e / P) * WW + rc + e % P];
  __syncthreads();

  for (int o = tid; o < WIN_ * WIN_; o += 256) {
    int dr = o / WIN_ - V_, dc = o % WIN_ - V_;
    int cr = rr + dr, cc = rc + dc;
    float d;
    if (cr < 0 || cr >= HP || cc < 0 || cc >= HP) {
      d = FINF;
    } else if (o == V_ * WIN_ + V_) {
      d = -FINF;  // self patch always selected first
    } else {
      float s = 0.f;
      for (int u = 0; u < P; ++u)
        for (int v = 0; v < P; ++v) {
          float t = img[(cr + u) * WW + cc + v] - ref[u * P + v] + EPS_;
          s += t * t;
        }
      d = __builtin_sqrtf(s);
    }
    dist[o] = d;
  }
  __syncthreads();

  for (int j = 0; j < K; ++j) {
    float best = FINF; int bo = 0;
    for (int o = tid; o < WIN_ * WIN_; o += 256)
      if (dist[o] < best) { best = dist[o]; bo = o; }
    rval[tid] = best; ridx[tid] = bo;
    __syncthreads();
    if (tid == 0) {
      float m = FINF; int mo = 0;
      for (int t = 0; t < 256; ++t)
        if (rval[t] < m) { m = rval[t]; mo = ridx[t]; }
      int dr = mo / WIN_ - V_, dc = mo % WIN_ - V_;
      int pr = imin(rr + dr, HP - 1), pc = imin(rc + dc, HP - 1);
      idx_out[b * K + j] = pr * HP + pc;
      dist[mo] = FINF;
    }
    __syncthreads();
  }
}

// ---------------------------------------------------------------------------
// Step-1 denoise: k=16, n=121. One wave per group.
//   G = Y*Y^T (WMMA), A = G + 0.25*n*s2*I, theta = A^-1 (G - n*s2*I),
//   Xhat = theta*Y (WMMA), aggregate with weights.
// ---------------------------------------------------------------------------
#define YS1 132  // >=128 columns, padded with zeros from 121
__global__ void __launch_bounds__(32)
denoise1_kernel(const float* __restrict__ yimg, const int* __restrict__ idx,
                const float* __restrict__ sigptr,
                float* __restrict__ numX, float* __restrict__ den) {
  __shared__ float Y[K1_][YS1];
  __shared__ float G[K1_][17];
  __shared__ float L[K1_][17];
  __shared__ float TH[K1_][17];
  __shared__ int   prow[K1_], pcol[K1_];
  __shared__ float w[K1_];

  const int b = blockIdx.x;
  const int tid = threadIdx.x;
  const int l = tid & 15, h = tid >> 4;

  // gather 16 patches of 121 components
  for (int e = tid; e < K1_ * N1_; e += 32) {
    int kap = e / N1_, c = e % N1_;
    int pi = idx[b * K1_ + kap];
    int pr = pi / HP1, pc = pi % HP1;
    Y[kap][c] = yimg[(pr + c / P1_) * WW + pc + c % P1_];
  }
  for (int e = tid; e < K1_ * (YS1 - N1_); e += 32) {
    int kap = e / (YS1 - N1_), c = N1_ + e % (YS1 - N1_);
    Y[kap][c] = 0.f;
  }
  if (tid < K1_) {
    int pi = idx[b * K1_ + tid];
    prow[tid] = pi / HP1; pcol[tid] = pi % HP1;
  }
  __syncthreads();

  // Gram: 31 WMMA steps over K=0..123 (pad zeros)
  {
    v8f acc = {};
    for (int kk = 0; kk < 124; kk += 4) {
      v2f a; a.x = Y[l][kk + 2 * h]; a.y = Y[l][kk + 2 * h + 1];
      acc = wmma_f32_4(a, a, acc);
    }
    for (int r = 0; r < 8; ++r) G[r + 8 * h][l] = acc[r];
  }
  __syncthreads();

  const float sg = *sigptr;
  const float s2 = sg * sg;
  const float lam  = (float)N1_ * 0.25f * s2;
  const float rhsd = (float)N1_ * s2;
  for (int e = tid; e < 256; e += 32) {
    int i = e >> 4, j = e & 15;
    float g = G[i][j];
    TH[i][j] = g - (i == j ? rhsd : 0.f);  // RHS
    L[i][j]  = g + (i == j ? lam  : 0.f);  // SPD system
  }
  __syncthreads();

  // Cholesky (16x16, right-looking)
  for (int k = 0; k < 16; ++k) {
    if (tid == 0) L[k][k] = __builtin_sqrtf(fmaxf(L[k][k], 1e-20f));
    __syncthreads();
    float dk = L[k][k];
    for (int i = k + 1 + tid; i < 16; i += 32) L[i][k] /= dk;
    __syncthreads();
    for (int j = k + 1; j < 16; ++j)
      for (int i = j + tid; i < 16; i += 32)
        L[i][j] -= L[i][k] * L[j][k];
    __syncthreads();
  }
  // column-parallel triangular solves: TH <- A^-1 * TH
  if (tid < 16) {
    int j = tid;
    for (int i = 0; i < 16; ++i) {
      float s = TH[i][j];
      for (int m = 0; m < i; ++m) s -= L[i][m] * TH[m][j];
      TH[i][j] = s / L[i][i];
    }
    for (int i = 15; i >= 0; --i) {
      float s = TH[i][j];
      for (int m = i + 1; m < 16; ++m) s -= L[m][i] * TH[m][j];
      TH[i][j] = s / L[i][i];
    }
  }
  __syncthreads();
  if (tid < 16) {
    float s = 0.f;
    for (int j = 0; j < 16; ++j) s += TH[tid][j] * TH[tid][j];
    s = fminf(fmaxf(s, 1.f / 16.f), 1.f);
    w[tid] = 1.f / s;
  }
  __syncthreads();

  // Xhat = theta * Y : 8 column tiles of 16, K=16 (4 WMMA steps each)
  for (int tj = 0; tj < 8; ++tj) {
    v8f acc = {};
    for (int kk = 0; kk < 16; kk += 4) {
      v2f a;  a.x = TH[l][kk + 2 * h];        a.y = TH[l][kk + 2 * h + 1];
      v2f bb; bb.x = Y[kk + 2 * h][16 * tj + l]; bb.y = Y[kk + 2 * h + 1][16 * tj + l];
      acc = wmma_f32_4(a, bb, acc);
    }
    int c = 16 * tj + l;
    if (c < N1_) {
      int u = c / P1_, v = c % P1_;
      for (int r = 0; r < 8; ++r) {
        int kap = r + 8 * h;
        float wk = w[kap];
        int px = (prow[kap] + u) * WW + (pcol[kap] + v);
        atomicAdd(&numX[px], wk * acc[r]);
        atomicAdd(&den[px], wk);
      }
    }
  }
}

// ---------------------------------------------------------------------------
// Step-2 denoise: k=64, n=36. One 256-thread block (8 waves) per group.
//   G = X*X^T (16 WMMA tiles), A = G + n*(t*s)^2 I, S = A^-1 G (chol+solve),
//   Xhat = S^T * Z (12 WMMA tiles), Zhat blend, aggregate.
// ---------------------------------------------------------------------------
#define ZS2 49  // >=48 cols, zero-padded from 36
__global__ void __launch_bounds__(256)
denoise2_kernel(const float* __restrict__ zimg, const float* __restrict__ ximg,
                const float* __restrict__ yimg, const int* __restrict__ idx,
                const float* __restrict__ sigptr, float tau,
                float* __restrict__ numX, float* __restrict__ numZ,
                float* __restrict__ den) {
  __shared__ float X[K2_][37];
  __shared__ float Z[K2_][ZS2];
  __shared__ float G[K2_ * 65];
  __shared__ float S[K2_ * 65];
  __shared__ int   prow[K2_], pcol[K2_];
  __shared__ float w[K2_];
  __shared__ float red[256], red2[256];
  __shared__ float sh_t, sh_sig;

  const int b = blockIdx.x;
  const int tid = threadIdx.x;
  const int wave = tid >> 5;
  const int l = tid & 15, h = (tid >> 4) & 1;

  // gather X,Z; accumulate sum/sumsq of (Y - Z) for the t statistic
  float psum = 0.f, psq = 0.f;
  for (int e = tid; e < K2_ * N2_; e += 256) {
    int kap = e / N2_, c = e % N2_;
    int pi = idx[b * K2_ + kap];
    int pr = pi / HP2, pc = pi % HP2;
    int px = (pr + c / P2_) * WW + pc + c % P2_;
    float zv = zimg[px];
    X[kap][c] = ximg[px];
    Z[kap][c] = zv;
    float d = yimg[px] - zv;
    psum += d; psq += d * d;
  }
  for (int e = tid; e < K2_ * (ZS2 - N2_); e += 256) {
    int kap = e / (ZS2 - N2_), c = N2_ + e % (ZS2 - N2_);
    Z[kap][c] = 0.f;
  }
  if (tid < K2_) {
    int pi = idx[b * K2_ + tid];
    prow[tid] = pi / HP2; pcol[tid] = pi % HP2;
  }
  red[tid] = psum; red2[tid] = psq;
  __syncthreads();
  if (tid == 0) {
    float s = 0.f, q = 0.f;
    for (int t = 0; t < 256; ++t) { s += red[t]; q += red2[t]; }
    const float inv = 1.f / (float)(K2_ * N2_);
    float mean = s * inv;
    float var = fmaxf(q * inv - mean * mean, 0.f);
    float sg = *sigptr;
    float t = 1.f - __builtin_sqrtf(var) / sg;
    sh_t = fmaxf(t, tau + 1e-6f);
    sh_sig = sg;
  }
  __syncthreads();

  // G = X * X^T : 16 tiles of 16x16, K=36 (9 WMMA steps each), 2 tiles/wave
  for (int tile = wave; tile < 16; tile += 8) {
    int ti = tile >> 2, tj = tile & 3;
    v8f acc = {};
    for (int kk = 0; kk < 36; kk += 4) {
      v2f a;  a.x = X[16 * ti + l][kk + 2 * h];  a.y = X[16 * ti + l][kk + 2 * h + 1];
      v2f bb; bb.x = X[16 * tj + l][kk + 2 * h]; bb.y = X[16 * tj + l][kk + 2 * h + 1];
      acc = wmma_f32_4(a, bb, acc);
    }
    for (int r = 0; r < 8; ++r)
      G[(16 * ti + r + 8 * h) * 65 + 16 * tj + l] = acc[r];
  }
  __syncthreads();

  // S <- G (RHS); G += lam*I then Cholesky in place
  const float ts = sh_t * sh_sig;
  const float lam = (float)N2_ * ts * ts;
  for (int e = tid; e < 64 * 64; e += 256) {
    int i = e >> 6, j = e & 63;
    S[i * 65 + j] = G[i * 65 + j];
    if (i == j) G[i * 65 + i] += lam;
  }
  __syncthreads();

  for (int k = 0; k < 64; ++k) {
    if (tid == 0) G[k * 65 + k] = __builtin_sqrtf(fmaxf(G[k * 65 + k], 1e-20f));
    __syncthreads();
    float dk = G[k * 65 + k];
    for (int i = k + 1 + tid; i < 64; i += 256) G[i * 65 + k] /= dk;
    __syncthreads();
    for (int j = k + 1; j < 64; ++j)
      for (int i = j + tid; i < 64; i += 256)
        G[i * 65 + j] -= G[i * 65 + k] * G[j * 65 + k];
    __syncthreads();
  }
  // column-parallel solves: S <- A^-1 * XtX  (xi = S^T)
  if (tid < 64) {
    int j = tid;
    for (int i = 0; i < 64; ++i) {
      float s = S[i * 65 + j];
      for (int m = 0; m < i; ++m) s -= G[i * 65 + m] * S[m * 65 + j];
      S[i * 65 + j] = s / G[i * 65 + i];
    }
    for (int i = 63; i >= 0; --i) {
      float s = S[i * 65 + j];
      for (int m = i + 1; m < 64; ++m) s -= G[m * 65 + i] * S[m * 65 + j];
      S[i * 65 + j] = s / G[i * 65 + i];
    }
  }
  __syncthreads();
  if (tid < 64) {  // weights: 1/max(col-sum of S^2, 1/k)
    float s = 0.f;
    for (int j = 0; j < 64; ++j) { float v = S[j * 65 + tid]; s += v * v; }
    w[tid] = 1.f / fmaxf(s, 1.f / 64.f);
  }
  __syncthreads();

  // Xhat = S^T * Z : 4x3 tiles, K=64 (16 WMMA steps each)
  const float ratio = tau / sh_t;
  const float fac = 1.f - ratio;
  for (int tile = wave; tile < 12; tile += 8) {
    int ti = tile / 3, tj = tile % 3;
    v8f acc = {};
    for (int kk = 0; kk < 64; kk += 4) {
      v2f a;  a.x = S[(kk + 2 * h) * 65 + 16 * ti + l];
              a.y = S[(kk + 2 * h + 1) * 65 + 16 * ti + l];
      v2f bb; bb.x = Z[kk + 2 * h][16 * tj + l];
              bb.y = Z[kk + 2 * h + 1][16 * tj + l];
      acc = wmma_f32_4(a, bb, acc);
    }
    int c = 16 * tj + l;
    if (c < N2_) {
      int u = c / P2_, v = c % P2_;
      for (int r = 0; r < 8; ++r) {
        int kap = 16 * ti + r + 8 * h;
        float xh = acc[r];
        float zh = fac * xh + ratio * Z[kap][c];
        float wk = w[kap];
        int px = (prow[kap] + u) * WW + (pcol[kap] + v);
        atomicAdd(&numX[px], wk * xh);
        atomicAdd(&numZ[px], wk * zh);
        atomicAdd(&den[px], wk);
      }
    }
  }
}

// ---------------------------------------------------------------------------
__global__ void __launch_bounds__(256)
zero_kernel(float* __restrict__ a, int n) {
  int i = blockIdx.x * 256 + threadIdx.x;
  if (i < n) a[i] = 0.f;
}

__global__ void __launch_bounds__(256)
normalize1_kernel(const float* __restrict__ num, const float* __restrict__ den,
                  float* __restrict__ out) {
  int i = blockIdx.x * 256 + threadIdx.x;
  if (i < NPIX) out[i] = num[i] / den[i];
}

__global__ void __launch_bounds__(256)
normalize2_kernel(const float* __restrict__ numX, const float* __restrict__ numZ,
                  const float* __restrict__ den,
                  float* __restrict__ xo, float* __restrict__ zo) {
  int i = blockIdx.x * 256 + threadIdx.x;
  if (i < NPIX) {
    float d = den[i];
    xo[i] = numX[i] / d;
    zo[i] = numZ[i] / d;
  }
}

// ---------------------------------------------------------------------------
extern "C" void kernel_launch(void* const* d_in, const int* in_sizes, int n_in,
                              void* d_out, int out_size, void* d_ws, size_t ws_size,
                              hipStream_t stream) {
  (void)in_sizes; (void)n_in; (void)out_size; (void)ws_size;
  const float* y   = (const float*)d_in[0];
  const float* sig = (const float*)d_in[1];

  float* ws   = (float*)d_ws;
  float* xA   = ws + 0 * NPIX;
  float* xB   = ws + 1 * NPIX;
  float* zA   = ws + 2 * NPIX;
  float* zB   = ws + 3 * NPIX;
  float* numX = ws + 4 * NPIX;   // numX,numZ,den contiguous for one zero pass
  float* numZ = ws + 5 * NPIX;
  float* den  = ws + 6 * NPIX;
  int* idx1 = (int*)(ws + 7 * NPIX);
  int* idx2 = idx1 + B1 * K1_;

  const int gz = (3 * NPIX + 255) / 256;
  const int gn = (NPIX + 255) / 256;

  // ---- step 1: x = denoise1(y) ----
  bm_kernel<P1_, K1_><<<B1, 256, 0, stream>>>(y, idx1, HR1);
  zero_kernel<<<gz, 256, 0, stream>>>(numX, 3 * NPIX);
  denoise1_kernel<<<B1, 32, 0, stream>>>(y, idx1, sig, numX, den);
  normalize1_kernel<<<gn, 256, 0, stream>>>(numX, den, xA);
  hipMemcpyAsync(zA, y, NPIX * sizeof(float), hipMemcpyDeviceToDevice, stream);

  // ---- 9 iterations of step 2 ----
  float* xc = xA; float* xn = xB;
  float* zc = zA; float* zn = zB;
  for (int m = 0; m < 9; ++m) {
    float tau = (1.f - (float)(m + 1) / 9.f) * 0.75f;
    if (m % 3 == 0)
      bm_kernel<P2_, K2_><<<B2, 256, 0, stream>>>(zc, idx2, HR2);
    zero_kernel<<<gz, 256, 0, stream>>>(numX, 3 * NPIX);
    denoise2_kernel<<<B2, 256, 0, stream>>>(zc, xc, y, idx2, sig, tau,
                                            numX, numZ, den);
    normalize2_kernel<<<gn, 256, 0, stream>>>(numX, numZ, den, xn, zn);
    float* t;
    t = xc; xc = xn; xn = t;
    t = zc; zc = zn; zn = t;
  }

  hipMemcpyAsync(d_out, zc, NPIX * sizeof(float), hipMemcpyDeviceToDevice, stream);
}